// GCN_8967891714351
// MI455X (gfx1250) — compile-verified
//
#include <hip/hip_runtime.h>
#include <hip/hip_bf16.h>

// GCN forward on MI455X (gfx1250, wave32):
//   support = x @ W            -> v_wmma_f32_16x16x4_f32, A staged in LDS (bank-padded)
//   agg     = scatter-add      -> L2-resident gather + global_atomic_add_f32
//   out     = log_softmax(relu(agg + b)) -> one wave32 per row, shfl_xor reductions

#define NFEAT 256
#define NHID  128
#define XS_STRIDE 260   // 256 + 4 pad: lane l hits bank (4*l + ka) % 64 -> conflict-free

typedef float v2f __attribute__((ext_vector_type(2)));
typedef float v8f __attribute__((ext_vector_type(8)));

// ---------------------------------------------------------------------------
// Kernel 1: init d_out[row][c] = bias[c]  (accumulator base for scatter-add)
// ---------------------------------------------------------------------------
__global__ __launch_bounds__(256) void gcn_init_bias(float4* __restrict__ out4,
                                                     const float4* __restrict__ bias4,
                                                     int nQuads) {  // nNodes * 32
    int gid = blockIdx.x * 256 + threadIdx.x;
    if (gid < nQuads) {
        out4[gid] = bias4[gid & 31];  // 128 floats = 32 float4 per row
    }
}

// ---------------------------------------------------------------------------
// Kernel 2: support[M,128] = x[M,256] @ W[256,128] via V_WMMA_F32_16X16X4_F32
// Block = 256 threads = 8 waves. The block's 16-row x-stripe (16 KB) is staged
// in LDS once (it is shared by all 8 waves); each wave owns one 16x16 output
// tile (wave id -> 16-col stripe, 8*16 = 128 = NHID).
//
// Fragment layout (ISA 7.12.2, 32-bit, wave32):
//   A 16x4 : lane l (0-15): M=l, {v0,v1}={K0,K1}; lane 16+l: M=l, {v0,v1}={K2,K3}
//   B 4x16 : lane l (0-15): N=l, {v0,v1}={K0,K1}; lane 16+l: N=l, {v0,v1}={K2,K3}
//   C/D    : VGPR j, lanes 0-15: (M=j, N=lane); lanes 16-31: (M=j+8, N=lane-16)
// ---------------------------------------------------------------------------
__global__ __launch_bounds__(256) void gcn_gemm_wmma(const float* __restrict__ X,
                                                     const float* __restrict__ W,
                                                     float* __restrict__ S,
                                                     int nRows) {
    __shared__ float xs[16 * XS_STRIDE];   // ~16.6 KB of the 320 KB WGP LDS

    const int lane = threadIdx.x & 31;
    const int wave = threadIdx.x >> 5;     // 0..7 -> column tile
    const int l    = lane & 15;
    const int half = lane >> 4;            // 0 or 1
    const int row0 = blockIdx.x * 16;
    const int col0 = wave * 16;

    // --- Stage the 16x256 x-stripe into LDS (coalesced float4, 4 per thread) ---
    #pragma unroll
    for (int i = 0; i < 4; ++i) {
        const int gid = threadIdx.x + i * 256;     // 0..1023 float4 slots
        const int r   = gid >> 6;                  // row in stripe (0..15)
        const int q   = gid & 63;                  // float4 within row
        const int rr  = min(row0 + r, nRows - 1);  // clamp ragged tail loads
        const float4 v = *(const float4*)(X + (size_t)rr * NFEAT + q * 4);
        *(float4*)(&xs[r * XS_STRIDE + q * 4]) = v;
    }
    __syncthreads();

    const float* __restrict__ xl = &xs[l * XS_STRIDE];

    v8f acc = {};
    #pragma unroll 8
    for (int k0 = 0; k0 < NFEAT; k0 += 4) {
        const int ka = k0 + 2 * half;              // even -> 8B-aligned LDS read
        v2f a = *(const v2f*)(xl + ka);
        v2f b;
        b.x = W[(size_t)ka * NHID + col0 + l];
        b.y = W[(size_t)(ka + 1) * NHID + col0 + l];
        acc = __builtin_amdgcn_wmma_f32_16x16x4_f32(
                  /*neg_a=*/false, a, /*neg_b=*/false, b,
                  /*c_mod=*/(short)0, acc, /*reuse_a=*/false, /*reuse_b=*/false);
    }

    // --- Epilogue: uniform branch, straight-line stores in the common case ---
    if (row0 + 16 <= nRows) {
        float* o = S + (size_t)(row0 + 8 * half) * NHID + col0 + l;
        #pragma unroll
        for (int j = 0; j < 8; ++j) {
            o[(size_t)j * NHID] = acc[j];
        }
    } else {
        #pragma unroll
        for (int j = 0; j < 8; ++j) {
            const int r = row0 + j + 8 * half;
            if (r < nRows) S[(size_t)r * NHID + col0 + l] = acc[j];
        }
    }
}

// ---------------------------------------------------------------------------
// Kernel 3: edge scatter. One wave per edge: 32 lanes x float4 = 128 features.
// support fits in L2 (25.6 MB << 192 MB) so gathers + atomics are L2-resident.
// ---------------------------------------------------------------------------
__global__ __launch_bounds__(256) void gcn_spmm_scatter(const int* __restrict__ src,
                                                        const int* __restrict__ dst,
                                                        const float* __restrict__ ew,
                                                        const float* __restrict__ S,
                                                        float* __restrict__ out,
                                                        int nEdges) {
    const int wave = threadIdx.x >> 5;
    const int lane = threadIdx.x & 31;
    const int e = blockIdx.x * 8 + wave;      // 8 edges per 256-thread block
    if (e >= nEdges) return;

    const int   s = src[e];
    const int   d = dst[e];
    const float w = ew[e];

    const float4 v = *(const float4*)(S + (size_t)s * NHID + lane * 4);
    float* o = out + (size_t)d * NHID + lane * 4;
    atomicAdd(o + 0, v.x * w);
    atomicAdd(o + 1, v.y * w);
    atomicAdd(o + 2, v.z * w);
    atomicAdd(o + 3, v.w * w);
}

// ---------------------------------------------------------------------------
// Kernel 4: in-place h = log_softmax(relu(h)) per row of 128. One wave per row.
// ---------------------------------------------------------------------------
__global__ __launch_bounds__(256) void gcn_relu_logsoftmax(float* __restrict__ out,
                                                           int nRows) {
    const int wave = (blockIdx.x * 256 + threadIdx.x) >> 5;
    const int lane = threadIdx.x & 31;
    if (wave >= nRows) return;

    float* row = out + (size_t)wave * NHID;
    float4 v = *(float4*)(row + lane * 4);
    v.x = fmaxf(v.x, 0.0f);
    v.y = fmaxf(v.y, 0.0f);
    v.z = fmaxf(v.z, 0.0f);
    v.w = fmaxf(v.w, 0.0f);

    // wave32 max-reduction
    float m = fmaxf(fmaxf(v.x, v.y), fmaxf(v.z, v.w));
    #pragma unroll
    for (int off = 16; off > 0; off >>= 1)
        m = fmaxf(m, __shfl_xor(m, off, 32));

    // wave32 sum of exp(h - m)
    float s = expf(v.x - m) + expf(v.y - m) + expf(v.z - m) + expf(v.w - m);
    #pragma unroll
    for (int off = 16; off > 0; off >>= 1)
        s += __shfl_xor(s, off, 32);

    const float lse = m + logf(s);
    v.x -= lse; v.y -= lse; v.z -= lse; v.w -= lse;
    *(float4*)(row + lane * 4) = v;
}

// ---------------------------------------------------------------------------
extern "C" void kernel_launch(void* const* d_in, const int* in_sizes, int n_in,
                              void* d_out, int out_size, void* d_ws, size_t ws_size,
                              hipStream_t stream) {
    const float* x      = (const float*)d_in[0];  // [N, 256]
    const float* weight = (const float*)d_in[1];  // [256, 128]
    const float* bias   = (const float*)d_in[2];  // [128]
    const int*   eidx   = (const int*)  d_in[3];  // [2, E]
    const float* ew     = (const float*)d_in[4];  // [E]

    const int nNodes = in_sizes[0] / NFEAT;       // 50000
    const int nEdges = in_sizes[4];               // 1600000
    const int* src = eidx;
    const int* dst = eidx + nEdges;

    float* out     = (float*)d_out;               // [N, 128] -> agg accumulator, then result
    float* support = (float*)d_ws;                // [N, 128] scratch (25.6 MB)

    // 1) out = bias (broadcast)
    {
        const int nQuads = nNodes * (NHID / 4);
        gcn_init_bias<<<(nQuads + 255) / 256, 256, 0, stream>>>(
            (float4*)out, (const float4*)bias, nQuads);
    }
    // 2) support = x @ W (WMMA f32, LDS-staged A)
    {
        const int rowTiles = (nNodes + 15) / 16;  // 3125
        gcn_gemm_wmma<<<rowTiles, 256, 0, stream>>>(x, weight, support, nNodes);
    }
    // 3) out += scatter(support[src] * ew)
    {
        const int blocks = (nEdges + 7) / 8;      // 8 edges (waves) per block
        gcn_spmm_scatter<<<blocks, 256, 0, stream>>>(src, dst, ew, support, out, nEdges);
    }
    // 4) out = log_softmax(relu(out)) in place
    {
        const int blocks = (nNodes + 7) / 8;      // 8 rows (waves) per block
        gcn_relu_logsoftmax<<<blocks, 256, 0, stream>>>(out, nNodes);
    }
}